// SCNNLight_41729902248132
// MI455X (gfx1250) — compile-verified
//
#include <hip/hip_runtime.h>

typedef float v2f __attribute__((ext_vector_type(2)));
typedef float v8f __attribute__((ext_vector_type(8)));
typedef float v4f __attribute__((ext_vector_type(4)));

__device__ __forceinline__ v8f wmma4(v2f a, v2f b, v8f c) {
  // D = A(16x4 f32) x B(4x16 f32) + C(16x16 f32), full-precision matrix pipe
  return __builtin_amdgcn_wmma_f32_16x16x4_f32(false, a, false, b, (short)0, c,
                                               false, false);
}

// ---------------- workspace layout (float offsets) ----------------
// Padded activation buffers (zero halos -> no bounds checks in conv loops).
#define F_FEATSP 0L                        // 8*68^3                = 2515456
#define F_Y0P    2515456L                  // 8*66^3*16             = 36799488
#define F_Y1P    (F_Y0P + 36799488L)       // 8*34^3*16             = 5030912
#define F_Y2P    (F_Y1P + 5030912L)        // 8*18^3*16             = 746496
#define F_Y3P    (F_Y2P + 746496L)         // 8*10^3*16             = 128000
#define F_Y4P    (F_Y3P + 128000L)         // 8*6^3*16              = 27648
#define F_Y5P    (F_Y4P + 27648L)          // 8*4^3*16              = 8192
#define F_Y6P    (F_Y5P + 8192L)           // 8*3^3*16              = 3456
#define F_ZEND   (F_Y6P + 3456L)           // = 45259648 (zeroed as float4)
#define F_M1     F_ZEND                    // 8*32^3                = 262144
#define F_M2     (F_M1 + 262144L)          // 8*16^3                = 32768
#define F_M3     (F_M2 + 32768L)           // 8*8^3                 = 4096
#define F_M4     (F_M3 + 4096L)            // 8*4^3                 = 512
#define F_M5     (F_M4 + 512L)             // 8*2^3                 = 64
#define F_M6     (F_M5 + 64L)              // 8
#define F_PSUM   (F_M6 + 8L)               // 6*8*16                = 768
#define F_MSUM   (F_PSUM + 768L)           // 6*8                   = 48
#define F_POOLS  (F_MSUM + 48L)            // 8*96                  = 768

// ---------------- zero / prep kernels ----------------
__global__ void k_zero4(float* __restrict__ ws) {   // zero all padded buffers
  long i = (long)blockIdx.x * blockDim.x + threadIdx.x;
  if (i < (F_ZEND >> 2)) ((v4f*)ws)[i] = (v4f){0.f, 0.f, 0.f, 0.f};
}

__global__ void k_zero(float* __restrict__ ws) {    // psum(768) + msum(48)
  int i = blockIdx.x * blockDim.x + threadIdx.x;
  if (i < 816) ws[F_PSUM + i] = 0.0f;
}

__global__ void k_feats(const float* __restrict__ x, const int* __restrict__ mask,
                        float* __restrict__ featsP) {
  int i = blockIdx.x * blockDim.x + threadIdx.x;
  if (i >= 8 * 64 * 64 * 64) return;
  int b = i >> 18, r = i & 262143;
  int iz = r >> 12, iy = (r >> 6) & 63, ix = r & 63;
  featsP[(((long)b * 68 + iz + 2) * 68 + iy + 2) * 68 + ix + 2] =
      x[i] * (float)mask[i];
}

// first mask pool: int mask 64^3 -> float 32^3 (2x2x2 max == OR for {0,1})
__global__ void k_pool_int(const int* __restrict__ mask, float* __restrict__ dst,
                           float* __restrict__ msum) {
  int i = blockIdx.x * blockDim.x + threadIdx.x;
  if (i >= 8 * 32768) return;
  int b = i >> 15, r = i & 32767;
  int oz = r >> 10, oy = (r >> 5) & 31, ox = r & 31;
  int v = 0;
  for (int dz = 0; dz < 2; dz++)
    for (int dy = 0; dy < 2; dy++)
      for (int dx = 0; dx < 2; dx++)
        v |= mask[(((long)b * 64 + 2 * oz + dz) * 64 + 2 * oy + dy) * 64 + 2 * ox + dx];
  float f = (v != 0) ? 1.0f : 0.0f;
  dst[i] = f;
  atomicAdd(&msum[b], f);   // exact: small-int float adds
}

__global__ void k_pool(const float* __restrict__ src, float* __restrict__ dst,
                       float* __restrict__ msum, int dD) {
  int dD3 = dD * dD * dD;
  int i = blockIdx.x * blockDim.x + threadIdx.x;
  if (i >= 8 * dD3) return;
  int b = i / dD3, r = i % dD3;
  int oz = r / (dD * dD), oy = (r / dD) % dD, ox = r % dD;
  int sD = 2 * dD;
  float v = 0.0f;
  for (int dz = 0; dz < 2; dz++)
    for (int dy = 0; dy < 2; dy++)
      for (int dx = 0; dx < 2; dx++) {
        float s = src[(((long)b * sD + 2 * oz + dz) * sD + 2 * oy + dy) * sD + 2 * ox + dx];
        v = fmaxf(v, s);
      }
  dst[i] = v;
  atomicAdd(&msum[b], v);
}

// ---------------- stem: conv 5x5x5, 1->16ch via WMMA ---------------------------
// Block = 8 waves covering 2 y-rows x 64 x. Input halo tile (5 x 6 x 68, padded
// to row stride 72) staged in LDS; K remapped to row*8+dx (25 rows, dx 5..7
// zero-weighted) so every A/B operand is an LDS load at a constant offset.
__global__ __launch_bounds__(256) void k_stem(const float* __restrict__ featsP,
                                              const int* __restrict__ mask,
                                              const float* __restrict__ wstem,
                                              float* __restrict__ y0P) {
  __shared__ float wl[3200];   // B: 25 rows x 8 dx x 16 ch (dx>=5 -> 0)
  __shared__ float ft[2160];   // A: [dz 0..4][py 0..5][px 0..71] (pad zeroed)

  int T0 = blockIdx.x * 8;
  int b = T0 >> 14, z = (T0 >> 8) & 63, ybase = (T0 >> 2) & 63;

  for (int i = threadIdx.x; i < 3200; i += 256) {
    int K = i >> 4, n = i & 15;
    int row = K >> 3, dx = K & 7;
    wl[i] = (dx < 5) ? wstem[(row * 5 + dx) * 16 + n] : 0.0f;
  }
  for (int i = threadIdx.x; i < 2160; i += 256) {
    int dz = i / 432, rem = i % 432;
    int py = rem / 72, px = rem % 72;
    int gy = ybase + py;                       // padded y index
    float v = 0.0f;
    if (px < 68 && gy < 68)
      v = featsP[(((long)b * 68 + z + dz) * 68 + gy) * 68 + px];
    ft[i] = v;
  }
  __syncthreads();

  int wave = threadIdx.x >> 5, lane = threadIdx.x & 31;
  int half = lane >> 4, ln = lane & 15;
  int yloc = wave >> 2, x0 = (wave & 3) << 4;
  int y = ybase + yloc;

  int ab = yloc * 72 + x0 + ln + 2 * half;     // per-lane A base (dx folded)
  int wb0 = 32 * half + ln;                    // per-lane B base

  v8f acc = {0.f, 0.f, 0.f, 0.f, 0.f, 0.f, 0.f, 0.f};
#pragma unroll
  for (int row = 0; row < 25; row++) {
    const int ro = ((row / 5) * 6 + (row % 5)) * 72;
#pragma unroll
    for (int c = 0; c < 2; c++) {
      v2f a = {ft[ab + ro + 4 * c], ft[ab + ro + 4 * c + 1]};
      const int kb = (row * 8 + 4 * c) * 16;
      v2f bb = {wl[wb0 + kb], wl[wb0 + kb + 16]};
      acc = wmma4(a, bb, acc);
    }
  }

#pragma unroll
  for (int v = 0; v < 8; v++) {
    int M = v + 8 * half;
    int xo = x0 + M;
    float mv = (float)mask[(((long)b * 64 + z) * 64 + y) * 64 + xo];
    float val = acc[v] * mv;
    val = val > 0.0f ? val : 0.0f;
    y0P[((((long)b * 66 + z + 1) * 66 + y + 1) * 66 + xo + 1) * 16 + ln] = val;
  }
}

// ---------------- stage conv: 3x3x3 stride 2, 16->16ch, K=432, WMMA ------------
// src padded (inD+2)^3, dst padded (outD+2)^3; compile-time dims. 2*half folded
// into A base -> every A operand is one aligned global_load_b64 at imm offset.
template <int inD, int outD>
__global__ __launch_bounds__(256) void k_stage(const float* __restrict__ src,
                                               float* __restrict__ dst,
                                               const float* __restrict__ pmask,
                                               const float* __restrict__ w,
                                               float* __restrict__ psum) {
  constexpr int pin = inD + 2, pout = outD + 2;
  constexpr int outD3 = outD * outD * outD;
  constexpr int nvox = 8 * outD3;
  constexpr int ntiles = (nvox + 15) / 16;
  constexpr bool fast = (outD3 % 16) == 0;           // tile within one batch

  __shared__ float wl[6912];                         // B matrix K=432 x N=16
  for (int i = threadIdx.x; i < 6912; i += 256) wl[i] = w[i];
  __syncthreads();

  int wave = threadIdx.x >> 5, lane = threadIdx.x & 31;
  int half = lane >> 4, ln = lane & 15;
  int T = blockIdx.x * 8 + wave;
  if (T >= ntiles) return;

  int g0 = T * 16;
  int gA = g0 + ln;
  bool va = gA < nvox;
  int gc = va ? gA : 0;
  int b = gc / outD3, r = gc % outD3;
  int oz = r / (outD * outD), oy = (r / outD) % outD, ox = r % outD;
  // padded input base at (2*oz, 2*oy, 2*ox); fold 2*half (cin pair select)
  const float* spb =
      src + ((((long)b * pin + 2 * oz) * pin + 2 * oy) * pin + 2 * ox) * 16 +
      2 * half;
  const int wb0 = 32 * half + ln;

  v8f acc = {0.f, 0.f, 0.f, 0.f, 0.f, 0.f, 0.f, 0.f};
#pragma unroll
  for (int t = 0; t < 27; t++) {
    const int dz = t / 9, dy = (t / 3) % 3, dx = t % 3;
    const int ro = ((dz * pin + dy) * pin + dx) * 16;
#pragma unroll
    for (int c = 0; c < 4; c++) {
      v2f a = *(const v2f*)(spb + ro + 4 * c);
      const int kb = (t * 16 + 4 * c) * 16;
      v2f bb = {wl[wb0 + kb], wl[wb0 + kb + 16]};
      acc = wmma4(a, bb, acc);
    }
  }

  float lsum = 0.0f;
  int col = ln;
#pragma unroll
  for (int v = 0; v < 8; v++) {
    int g = g0 + v + 8 * half;
    if (g < nvox) {
      int bv = g / outD3, rv = g % outD3;
      int vz = rv / (outD * outD), vy = (rv / outD) % outD, vx = rv % outD;
      float mv = pmask[g];
      float val = acc[v] * mv;
      val = val > 0.0f ? val : 0.0f;
      dst[((((long)bv * pout + vz + 1) * pout + vy + 1) * pout + vx + 1) * 16 + col] = val;
      if (fast) lsum += val;
      else atomicAdd(&psum[(long)bv * 16 + col], val);
    }
  }
  if (fast) {
    float tot = lsum + __shfl_xor(lsum, 16, 32);     // combine lane halves
    if (half == 0) atomicAdd(&psum[(long)b * 16 + col], tot);
  }
}

// ---------------- pooled-feature normalize ----------------
__global__ void k_pools(const float* __restrict__ psum, const float* __restrict__ msum,
                        float* __restrict__ pools) {
  int i = blockIdx.x * blockDim.x + threadIdx.x;
  if (i >= 768) return;
  int b = i / 96, r = i % 96, s = r / 16, c = r % 16;
  float d = fmaxf(msum[s * 8 + b], 1.0f);
  pools[i] = psum[(s * 8 + b) * 16 + c] / d;
}

// ---------------- head: all dense layers in one block, the 128-wide ones via WMMA ----------------
__global__ __launch_bounds__(256) void k_head(
    const float* __restrict__ pools, const float* __restrict__ meta,
    const float* __restrict__ sched, const float* __restrict__ mw1,
    const float* __restrict__ mb1, const float* __restrict__ mw2,
    const float* __restrict__ mb2, const float* __restrict__ matw,
    const float* __restrict__ matb, const float* __restrict__ schw,
    const float* __restrict__ schb, const float* __restrict__ f1w,
    const float* __restrict__ f1b, const float* __restrict__ f2w,
    const float* __restrict__ f2b, float* __restrict__ out) {
  __shared__ float H[8 * 32];
  __shared__ float A1[16 * 112];
  __shared__ float Z[16 * 256];
  __shared__ float ZR[16 * 128];
  int tid = threadIdx.x;

  {  // meta hidden layer (8x32)
    int b = tid >> 5, j = tid & 31;
    float s = mb1[j];
    for (int k = 0; k < 3; k++) s += meta[b * 3 + k] * mw1[k * 32 + j];
    H[tid] = fmaxf(s, 0.0f);
  }
  __syncthreads();

  // A1 = [pools | meta_emb], rows 8..15 zero-padded
  for (int i = tid; i < 16 * 112; i += 256) {
    int r = i / 112, c = i % 112;
    float v = 0.0f;
    if (r < 8) {
      if (c < 96) v = pools[r * 96 + c];
      else {
        int j = c - 96;
        float s = mb2[j];
        for (int k = 0; k < 32; k++) s += H[r * 32 + k] * mw2[k * 16 + j];
        v = s;
      }
    }
    A1[i] = v;
  }
  __syncthreads();

  int wave = tid >> 5, lane = tid & 31, half = lane >> 4, ln = lane & 15;
  int col = wave * 16 + ln;                          // N column, 8 waves x 16

  {  // m_emb = relu(A1 @ mat_w + mat_b) -> Z[:,0:128]
    v8f acc = {0.f, 0.f, 0.f, 0.f, 0.f, 0.f, 0.f, 0.f};
#pragma unroll
    for (int kk = 0; kk < 28; kk++) {
      int kb = kk * 4 + 2 * half;
      v2f a = *(const v2f*)&A1[ln * 112 + kb];
      v2f bb = {matw[kb * 128 + col], matw[(kb + 1) * 128 + col]};
      acc = wmma4(a, bb, acc);
    }
    float bias = matb[col];
#pragma unroll
    for (int v = 0; v < 8; v++) {
      int rr = v + 8 * half;
      Z[rr * 256 + col] = fmaxf(acc[v] + bias, 0.0f);
    }
  }
  {  // s_emb = relu(sched @ sch_w + sch_b) -> Z[:,128:256]
    v8f acc = {0.f, 0.f, 0.f, 0.f, 0.f, 0.f, 0.f, 0.f};
#pragma unroll
    for (int kk = 0; kk < 32; kk++) {
      int kb = kk * 4 + 2 * half;
      v2f a = {0.0f, 0.0f};
      if (ln < 8) a = *(const v2f*)&sched[ln * 128 + kb];
      v2f bb = {schw[kb * 128 + col], schw[(kb + 1) * 128 + col]};
      acc = wmma4(a, bb, acc);
    }
    float bias = schb[col];
#pragma unroll
    for (int v = 0; v < 8; v++) {
      int rr = v + 8 * half;
      Z[rr * 256 + 128 + col] = fmaxf(acc[v] + bias, 0.0f);
    }
  }
  __syncthreads();

  {  // relu(Z @ f1_w + f1_b) -> ZR
    v8f acc = {0.f, 0.f, 0.f, 0.f, 0.f, 0.f, 0.f, 0.f};
#pragma unroll
    for (int kk = 0; kk < 64; kk++) {
      int kb = kk * 4 + 2 * half;
      v2f a = *(const v2f*)&Z[ln * 256 + kb];
      v2f bb = {f1w[kb * 128 + col], f1w[(kb + 1) * 128 + col]};
      acc = wmma4(a, bb, acc);
    }
    float bias = f1b[col];
#pragma unroll
    for (int v = 0; v < 8; v++) {
      int rr = v + 8 * half;
      ZR[rr * 128 + col] = fmaxf(acc[v] + bias, 0.0f);
    }
  }
  __syncthreads();

  if (tid < 8) {  // final 128 -> 1
    float s = f2b[0];
    for (int k = 0; k < 128; k++) s += ZR[tid * 128 + k] * f2w[k];
    out[tid] = s;
  }
}

// ---------------- launcher ----------------
extern "C" void kernel_launch(void* const* d_in, const int* in_sizes, int n_in,
                              void* d_out, int out_size, void* d_ws, size_t ws_size,
                              hipStream_t stream) {
  (void)in_sizes; (void)n_in; (void)out_size; (void)ws_size;
  const float* x     = (const float*)d_in[0];
  const int*   mask  = (const int*)d_in[1];
  const float* meta  = (const float*)d_in[2];
  const float* sched = (const float*)d_in[3];
  const float* wstem = (const float*)d_in[4];
  const float* wstg  = (const float*)d_in[5];
  const float* mw1   = (const float*)d_in[6];
  const float* mb1   = (const float*)d_in[7];
  const float* mw2   = (const float*)d_in[8];
  const float* mb2   = (const float*)d_in[9];
  const float* matw  = (const float*)d_in[10];
  const float* matb  = (const float*)d_in[11];
  const float* schw  = (const float*)d_in[12];
  const float* schb  = (const float*)d_in[13];
  const float* f1w   = (const float*)d_in[14];
  const float* f1b   = (const float*)d_in[15];
  const float* f2w   = (const float*)d_in[16];
  const float* f2b   = (const float*)d_in[17];

  float* ws     = (float*)d_ws;
  float* featsP = ws + F_FEATSP;
  float* psum   = ws + F_PSUM;
  float* msum   = ws + F_MSUM;
  float* pools  = ws + F_POOLS;
  float* y0P = ws + F_Y0P; float* y1P = ws + F_Y1P; float* y2P = ws + F_Y2P;
  float* y3P = ws + F_Y3P; float* y4P = ws + F_Y4P; float* y5P = ws + F_Y5P;
  float* y6P = ws + F_Y6P;
  float* m1 = ws + F_M1; float* m2 = ws + F_M2; float* m3 = ws + F_M3;
  float* m4 = ws + F_M4; float* m5 = ws + F_M5; float* m6 = ws + F_M6;

  {  // zero padded activation buffers (halos must be 0) + accumulators
    long n4 = F_ZEND >> 2;
    int blocks = (int)((n4 + 255) / 256);
    k_zero4<<<blocks, 256, 0, stream>>>(ws);
    k_zero<<<4, 256, 0, stream>>>(ws);
  }
  k_feats<<<8192, 256, 0, stream>>>(x, mask, featsP);

  k_pool_int<<<1024, 256, 0, stream>>>(mask, m1, msum + 0);
  k_pool<<<128, 256, 0, stream>>>(m1, m2, msum + 8, 16);
  k_pool<<<16, 256, 0, stream>>>(m2, m3, msum + 16, 8);
  k_pool<<<2, 256, 0, stream>>>(m3, m4, msum + 24, 4);
  k_pool<<<1, 256, 0, stream>>>(m4, m5, msum + 32, 2);
  k_pool<<<1, 256, 0, stream>>>(m5, m6, msum + 40, 1);

  k_stem<<<16384, 256, 0, stream>>>(featsP, mask, wstem, y0P);

  k_stage<64, 32><<<2048, 256, 0, stream>>>(y0P, y1P, m1, wstg + 0 * 6912, psum + 0 * 128);
  k_stage<32, 16><<<256, 256, 0, stream>>>(y1P, y2P, m2, wstg + 1 * 6912, psum + 1 * 128);
  k_stage<16, 8><<<32, 256, 0, stream>>>(y2P, y3P, m3, wstg + 2 * 6912, psum + 2 * 128);
  k_stage<8, 4><<<4, 256, 0, stream>>>(y3P, y4P, m4, wstg + 3 * 6912, psum + 3 * 128);
  k_stage<4, 2><<<1, 256, 0, stream>>>(y4P, y5P, m5, wstg + 4 * 6912, psum + 4 * 128);
  k_stage<2, 1><<<1, 256, 0, stream>>>(y5P, y6P, m6, wstg + 5 * 6912, psum + 5 * 128);

  k_pools<<<3, 256, 0, stream>>>(psum, msum, pools);
  k_head<<<1, 256, 0, stream>>>(pools, meta, sched, mw1, mb1, mw2, mb2, matw, matb,
                                schw, schb, f1w, f1b, f2w, f2b, (float*)d_out);
}